// SGNS_27376121545328
// MI455X (gfx1250) — compile-verified
//
#include <hip/hip_runtime.h>

// SGNS forward on gfx1250 (MI455X), wave32.
// One wave handles 16 batch elements via the diagonal of a 16x16 fp32 WMMA
// accumulator: D = P1(16x128) x P2^T(128x16), built from 32 chained
// V_WMMA_F32_16X16X4_F32 ops (exact fp32, matches reference precision).
// Gather addresses are strength-reduced to loop-carried pointers so the
// scattered global_load_b32s issue without 64-bit multiply chains.

typedef __attribute__((ext_vector_type(2))) float v2f;
typedef __attribute__((ext_vector_type(8))) float v8f;

#define W_DIM 128
#define BLOCK_THREADS 128   // 4 waves/block, 16 batch elems per wave -> 64/block

__global__ __launch_bounds__(BLOCK_THREADS) void sgns_wmma_f32_kernel(
    const int* __restrict__ idx1, const int* __restrict__ idx2,
    const float* __restrict__ W1, const float* __restrict__ b1,
    const float* __restrict__ W2, const float* __restrict__ b2,
    float* __restrict__ out, int D, int nBatch)
{
    __shared__ float sb1[W_DIM];
    __shared__ float sb2[W_DIM];

    const int t = threadIdx.x;
    // Stage biases in LDS once (128 threads == 128 bias entries).
    sb1[t] = b1[t];
    sb2[t] = b2[t];
    __syncthreads();

    const int lane = t & 31;
    const int wave = t >> 5;
    const int b0   = (blockIdx.x * (BLOCK_THREADS / 32) + wave) * 16;
    if (b0 >= nBatch) return;   // wave-uniform exit (EXEC stays all-ones)

    const int m  = lane & 15;   // A row (batch) / B column (batch)
    const int kh = lane >> 4;   // which K-half this half-wave supplies

    const size_t rowStep = (size_t)D;          // stride between k-rows (elems)
    const size_t c1 = (size_t)idx1[b0 + m];    // gathered column of W1
    const size_t c2 = (size_t)idx2[b0 + m];    // gathered column of W2

    // Loop-carried gather pointers: k = 2*kh and k = 2*kh+1 rows.
    const float* pA0 = W1 + (size_t)(2 * kh) * rowStep + c1;
    const float* pA1 = pA0 + rowStep;
    const float* pB0 = W2 + (size_t)(2 * kh) * rowStep + c2;
    const float* pB1 = pB0 + rowStep;
    const size_t step = 4 * rowStep;           // advance 4 k-rows per chunk

    v8f acc = {};   // 16x16 fp32 accumulator; we only consume the diagonal.

#pragma unroll
    for (int c = 0; c < W_DIM / 4; ++c) {
        const int k0 = 4 * c + 2 * kh;   // this lane supplies K = k0, k0+1
        v2f a, bm;
        // A: p1[m, k] = W1[k, idx1] + b1[k]   (16x4 fp32 tile)
        a.x  = __builtin_nontemporal_load(pA0) + sb1[k0    ];
        a.y  = __builtin_nontemporal_load(pA1) + sb1[k0 + 1];
        // B: p2[k, n] = W2[k, idx2] + b2[k]   (4x16 fp32 tile)
        bm.x = __builtin_nontemporal_load(pB0) + sb2[k0    ];
        bm.y = __builtin_nontemporal_load(pB1) + sb2[k0 + 1];
        // v_wmma_f32_16x16x4_f32: acc += A x B  (full fp32)
        acc = __builtin_amdgcn_wmma_f32_16x16x4_f32(
            /*neg_a=*/false, a, /*neg_b=*/false, bm,
            /*c_mod=*/(short)0, acc, /*reuse_a=*/false, /*reuse_b=*/false);
        pA0 += step; pA1 += step; pB0 += step; pB1 += step;
    }

    // Diagonal of C: lanes 0..7 -> batch m=lane (VGPR lane);
    //                lanes 24..31 -> batch m=lane-16 (VGPR lane-24).
    const bool lo = (lane < 8);
    const bool hi = (lane >= 24);
    if (lo || hi) {
        const int di   = lo ? lane : (lane - 24);
        const int bOut = b0 + (lo ? lane : (lane - 16));
        float dot = 0.0f;
#pragma unroll
        for (int r = 0; r < 8; ++r)      // v_cndmask chain, no scratch
            dot = (di == r) ? acc[r] : dot;
        const float s = 1.0f / (1.0f + __expf(-dot));
        reinterpret_cast<float2*>(out)[bOut] = make_float2(1.0f - s, s);
    }
}

extern "C" void kernel_launch(void* const* d_in, const int* in_sizes, int n_in,
                              void* d_out, int out_size, void* d_ws, size_t ws_size,
                              hipStream_t stream) {
    const int*   idx1 = (const int*)d_in[0];
    const int*   idx2 = (const int*)d_in[1];
    const float* W1   = (const float*)d_in[2];
    const float* b1   = (const float*)d_in[3];
    const float* W2   = (const float*)d_in[4];
    const float* b2   = (const float*)d_in[5];
    float*       out  = (float*)d_out;

    const int nBatch = in_sizes[0];            // 4096
    const int D      = in_sizes[2] / W_DIM;    // 100000

    const int batchPerBlock = (BLOCK_THREADS / 32) * 16;   // 64
    const int blocks = (nBatch + batchPerBlock - 1) / batchPerBlock;

    sgns_wmma_f32_kernel<<<blocks, BLOCK_THREADS, 0, stream>>>(
        idx1, idx2, W1, b1, W2, b2, out, D, nBatch);
}